// GraphConstrainedAttentionLayer_58188216926417
// MI455X (gfx1250) — compile-verified
//
#include <hip/hip_runtime.h>

typedef __attribute__((ext_vector_type(2))) float v2f;
typedef __attribute__((ext_vector_type(8))) float v8f;

#define DD   256   // model dim
#define HH   8     // heads
#define HDIM 32    // head dim
#define FEAT 64    // edge feature dim
#define HSTR 260   // LDS row stride (260 % 64 != 0 -> no bank conflict on column walks)

// ---------------------------------------------------------------------------
// f32 WMMA: D = A(16x4) * B(4x16) + C(16x16), full fp32 precision.
// ---------------------------------------------------------------------------
__device__ __forceinline__ v8f wmma4(v2f a, v2f b, v8f c) {
  return __builtin_amdgcn_wmma_f32_16x16x4_f32(false, a, false, b, (short)0, c,
                                               false, false);
}

// order-preserving float -> uint map for atomicMax-based segment max
__device__ __forceinline__ unsigned fmap(float x) {
  unsigned b = __float_as_uint(x);
  return (b & 0x80000000u) ? ~b : (b | 0x80000000u);
}
__device__ __forceinline__ float fdecode(unsigned u) {
  unsigned b = (u & 0x80000000u) ? (u ^ 0x80000000u) : ~u;
  return __uint_as_float(b);
}

// ---------------------------------------------------------------------------
__global__ __launch_bounds__(256) void zero_kernel(float* p, long n) {
  long i = (long)blockIdx.x * blockDim.x + threadIdx.x;
  if (i < n) p[i] = 0.0f;
}

// ---------------------------------------------------------------------------
// fp32 WMMA GEMM: C[M x Nd] = A[M x Kd] @ B[Kd x Nd] (+ bias)
// one wave per 16x32 output strip: A fragments (b64 loads) reused by two
// column tiles; 4 independent accumulator chains hide the WMMA RAW hazard.
// ---------------------------------------------------------------------------
template <bool BIAS>
__global__ __launch_bounds__(32) void gemm16x32(const float* __restrict__ A,
                                                const float* __restrict__ B,
                                                const float* __restrict__ bias,
                                                float* __restrict__ C,
                                                int M, int Kd, int Nd) {
  const int lane = threadIdx.x;
  const int row0 = blockIdx.x * 16;
  const int col0 = blockIdx.y * 32;               // two adjacent 16-wide tiles
  const int r    = lane & 15;
  const int ko   = (lane >> 4) << 1;              // 0 or 2 (K offset per half-wave)

  int rowA = row0 + r; if (rowA > M - 1) rowA = M - 1;   // clamp loads only
  const float* Arow = A + (long)rowA * Kd + ko;          // 8B aligned (k+ko even)
  const float* B0   = B + (long)ko * Nd + col0 + r;
  const float* B1   = B0 + 16;

  v8f acc00 = {}, acc01 = {}, acc10 = {}, acc11 = {};
  for (int k = 0; k < Kd; k += 8) {
    v2f a0 = *(const v2f*)(Arow + k);           // global_load_b64
    v2f a1 = *(const v2f*)(Arow + k + 4);
    v2f b00, b01, b10, b11;
    b00.x = B0[(long)k * Nd];       b00.y = B0[(long)(k + 1) * Nd];
    b10.x = B1[(long)k * Nd];       b10.y = B1[(long)(k + 1) * Nd];
    b01.x = B0[(long)(k + 4) * Nd]; b01.y = B0[(long)(k + 5) * Nd];
    b11.x = B1[(long)(k + 4) * Nd]; b11.y = B1[(long)(k + 5) * Nd];
    acc00 = wmma4(a0, b00, acc00);
    acc10 = wmma4(a0, b10, acc10);
    acc01 = wmma4(a1, b01, acc01);
    acc11 = wmma4(a1, b11, acc11);
  }

  const int colL    = lane & 15;
  const int rowBase = row0 + ((lane >> 4) << 3);
  float bv0 = BIAS ? bias[col0 + colL]      : 0.0f;
  float bv1 = BIAS ? bias[col0 + 16 + colL] : 0.0f;
#pragma unroll
  for (int i = 0; i < 8; ++i) {
    int row = rowBase + i;
    if (row < M) {
      C[(long)row * Nd + col0 + colL]      = acc00[i] + acc01[i] + bv0;
      C[(long)row * Nd + col0 + 16 + colL] = acc10[i] + acc11[i] + bv1;
    }
  }
}

// ---------------------------------------------------------------------------
// Fused edge kernel: one wave handles a 16-edge tile.
//  stage A: hid = relu(efeat_tile @ W1 + b1)   (WMMA, 16x64 @ 64x256 -> LDS)
//  stage B: mlp = hid @ W2 + b2                (VALU, 256x8, tiny)
//  stage C: score = QK/sqrt(hd) + mlp; store S; atomicMax segment max
// ---------------------------------------------------------------------------
__global__ __launch_bounds__(32) void edge_score_kernel(
    const float* __restrict__ efeat, const int* __restrict__ src,
    const int* __restrict__ dst, const float* __restrict__ Qm,
    const float* __restrict__ Km, const float* __restrict__ W1,
    const float* __restrict__ b1, const float* __restrict__ W2,
    const float* __restrict__ b2, float* __restrict__ S,
    unsigned* __restrict__ m_u, int E) {
  __shared__ float hid[16 * HSTR];
  __shared__ float sc[16 * HH];

  const int lane = threadIdx.x;
  const int e0   = blockIdx.x * 16;
  const int r    = lane & 15;
  const int ko   = (lane >> 4) << 1;

  int erow = e0 + r; if (erow > E - 1) erow = E - 1;

  // preload A-fragments of the 16x64 efeat tile (b64 loads; reused 16x)
  v2f af[16];
#pragma unroll
  for (int t = 0; t < 16; ++t)
    af[t] = *(const v2f*)(efeat + (long)erow * FEAT + t * 4 + ko);

  // ---- stage A: hidden = relu(efeat @ W1 + b1) -> LDS -------------------
  for (int ct = 0; ct < 16; ++ct) {
    const int col0 = ct * 16;
    const float* Bcol = W1 + (long)ko * DD + col0 + r;
    v8f acc0 = {}; v8f acc1 = {};
#pragma unroll
    for (int t = 0; t < 16; t += 2) {
      v2f b0, b1v;
      b0.x  = Bcol[(long)(t * 4) * DD];     b0.y  = Bcol[(long)(t * 4 + 1) * DD];
      b1v.x = Bcol[(long)(t * 4 + 4) * DD]; b1v.y = Bcol[(long)(t * 4 + 5) * DD];
      acc0 = wmma4(af[t],     b0,  acc0);
      acc1 = wmma4(af[t + 1], b1v, acc1);
    }
    const float bb      = b1[col0 + (lane & 15)];
    const int   rowBase = (lane >> 4) << 3;
#pragma unroll
    for (int i = 0; i < 8; ++i) {
      float v = acc0[i] + acc1[i] + bb;
      hid[(rowBase + i) * HSTR + col0 + (lane & 15)] = fmaxf(v, 0.0f);
    }
  }
  __syncthreads();

  // ---- stage B: mlp = hid @ W2 + b2  (16 edges x 8 heads = 128 outputs) --
#pragma unroll
  for (int i = 0; i < 4; ++i) {
    int p = lane + 32 * i;            // 0..127
    int e = p >> 3, hh = p & 7;
    float s = b2[hh];
    for (int d = 0; d < DD; ++d) s += hid[e * HSTR + d] * W2[d * HH + hh];
    sc[p] = s;
  }
  __syncthreads();

  // ---- stage C: QK dots + combine + segment max --------------------------
  const float scale = 0.17677669529663687f;  // 1/sqrt(32)
  for (int e = 0; e < 16; ++e) {
    if (e0 + e >= E) break;
    const int sn = src[e0 + e];
    const int dn = dst[e0 + e];
    const float* qrow = Qm + (long)dn * DD;   // Q[dst]
    const float* krow = Km + (long)sn * DD;   // K[src]
#pragma unroll
    for (int c = 0; c < HH; ++c) {            // head c == dims [c*32, c*32+31]
      float part = qrow[c * HDIM + lane] * krow[c * HDIM + lane];
#pragma unroll
      for (int off = 16; off; off >>= 1) part += __shfl_xor(part, off, 32);
      if (lane == 0) {
        float tot = part * scale + sc[e * HH + c];
        S[(long)(e0 + e) * HH + c] = tot;
        atomicMax(&m_u[(long)dn * HH + c], fmap(tot));
      }
    }
  }
}

// ---------------------------------------------------------------------------
__global__ __launch_bounds__(256) void softmax_exp_kernel(
    float* __restrict__ S, const int* __restrict__ dst,
    const unsigned* __restrict__ m_u, float* __restrict__ denom, long EH) {
  long i = (long)blockIdx.x * blockDim.x + threadIdx.x;
  if (i >= EH) return;
  int e  = (int)(i >> 3);
  int hh = (int)(i & 7);
  int dn = dst[e];
  float mval = fdecode(m_u[(long)dn * HH + hh]);
  float x = __expf(S[i] - mval);
  S[i] = x;
  atomicAdd(&denom[(long)dn * HH + hh], x);
}

// ---------------------------------------------------------------------------
// scatter: attn[dst] += alpha * V[src]; one wave per edge, lane d = c*32+lane
// ---------------------------------------------------------------------------
__global__ __launch_bounds__(256) void scatter_kernel(
    const float* __restrict__ S, const float* __restrict__ denom,
    const int* __restrict__ src, const int* __restrict__ dst,
    const float* __restrict__ Vm, float* __restrict__ attn, int E) {
  const int wave = threadIdx.x >> 5;
  const int lane = threadIdx.x & 31;
  long e = (long)blockIdx.x * 8 + wave;
  if (e >= E) return;
  const int sn = src[e];
  const int dn = dst[e];
  float alpha[HH];
#pragma unroll
  for (int hh = 0; hh < HH; ++hh)
    alpha[hh] = S[e * HH + hh] / denom[(long)dn * HH + hh];
  const float* vrow = Vm + (long)sn * DD;
  float* orow = attn + (long)dn * DD;
#pragma unroll
  for (int c = 0; c < 8; ++c) {
    int d = c * HDIM + lane;                // head index == c
    atomicAdd(&orow[d], alpha[c] * vrow[d]);
  }
}

// ---------------------------------------------------------------------------
extern "C" void kernel_launch(void* const* d_in, const int* in_sizes, int n_in,
                              void* d_out, int out_size, void* d_ws,
                              size_t ws_size, hipStream_t stream) {
  const float* h     = (const float*)d_in[0];
  const float* efeat = (const float*)d_in[1];
  const int*   src   = (const int*)d_in[2];
  const int*   dst   = (const int*)d_in[3];
  const float* Wq    = (const float*)d_in[4];
  const float* Wk    = (const float*)d_in[5];
  const float* Wv    = (const float*)d_in[6];
  const float* W1    = (const float*)d_in[7];
  const float* b1    = (const float*)d_in[8];
  const float* W2    = (const float*)d_in[9];
  const float* b2    = (const float*)d_in[10];
  const float* Wo    = (const float*)d_in[11];
  const float* bo    = (const float*)d_in[12];
  float* out = (float*)d_out;

  const int N = in_sizes[0] / DD;   // 50000
  const int E = in_sizes[2];        // 800000
  const long ND = (long)N * DD;

  // workspace layout (floats): Q | K | V | attn | denom | m_u | S
  float*    ws    = (float*)d_ws;
  float*    Q     = ws;
  float*    Kp    = Q + ND;
  float*    Vp    = Kp + ND;
  float*    attn  = Vp + ND;
  float*    denom = attn + ND;
  unsigned* m_u   = (unsigned*)(denom + (long)N * HH);
  float*    S     = (float*)((unsigned*)m_u + (long)N * HH);

  // zero attn + denom + m_u (contiguous)
  long zc = ND + 2L * N * HH;
  zero_kernel<<<(int)((zc + 255) / 256), 256, 0, stream>>>(attn, zc);

  // Q/K/V projections (fp32 WMMA, 16x32 strips)
  dim3 gqkv((N + 15) / 16, DD / 32);
  gemm16x32<false><<<gqkv, 32, 0, stream>>>(h, Wq, nullptr, Q,  N, DD, DD);
  gemm16x32<false><<<gqkv, 32, 0, stream>>>(h, Wk, nullptr, Kp, N, DD, DD);
  gemm16x32<false><<<gqkv, 32, 0, stream>>>(h, Wv, nullptr, Vp, N, DD, DD);

  // fused edge MLP + QK score + segment max
  edge_score_kernel<<<(E + 15) / 16, 32, 0, stream>>>(
      efeat, src, dst, Q, Kp, W1, b1, W2, b2, S, m_u, E);

  // exp + denominator
  long EH = (long)E * HH;
  softmax_exp_kernel<<<(int)((EH + 255) / 256), 256, 0, stream>>>(S, dst, m_u,
                                                                  denom, EH);

  // weighted scatter of V into attn
  scatter_kernel<<<(E + 7) / 8, 256, 0, stream>>>(S, denom, src, dst, Vp, attn,
                                                  E);

  // output projection (+bias)
  gemm16x32<true><<<gqkv, 32, 0, stream>>>(attn, Wo, bo, out, N, DD, DD);
}